// AcidSynthLearnedBiquadCoeff_62637803045380
// MI455X (gfx1250) — compile-verified
//
#include <hip/hip_runtime.h>
#include <math.h>

// ---------------------------------------------------------------------------
// AcidSynth on MI455X (gfx1250, wave32)
//   K_interp : coefficient lin-interp as banded matmul via v_wmma_f32_16x16x4_f32
//   K_phase1 : VCO + envelope + per-chunk affine IIR transition (T, c)
//   K_phase2 : lane-per-voice serial composition of chunk affine maps (1 wave)
//   K_phase3 : exact per-chunk IIR replay + fused FIR + tanh distortion
// ---------------------------------------------------------------------------

typedef __attribute__((ext_vector_type(2))) float v2f;
typedef __attribute__((ext_vector_type(8))) float v8f;

#define BSZ     32
#define NS      65536
#define NF      128
#define CHUNK   256
#define NCHUNK  256
#define STAB    0.999f
#define POS_SCALE (127.0f/65535.0f)
#define TWO_PI_F  6.2831853071795864769f
#define HALF_PI_F 1.5707963267948966192f

// output offsets (floats), tuple order: dry, wet, env, y_ab, a_coeff3, b_coeff, y_a
#define OFF_DRY  ((size_t)0)
#define OFF_WET  ((size_t)BSZ*NS)
#define OFF_ENV  ((size_t)2*BSZ*NS)
#define OFF_YAB  ((size_t)3*BSZ*NS)
#define OFF_A3   ((size_t)4*BSZ*NS)
#define OFF_B3   (OFF_A3 + (size_t)3*BSZ*NS)
#define OFF_YA   (OFF_B3 + (size_t)3*BSZ*NS)

// workspace offsets (floats)
#define WS_TRANS 0                       // [NCHUNK][BSZ][6]: T00 T01 T10 T11 c0 c1
#define WS_INIT  (NCHUNK*BSZ*6)         // [NCHUNK][BSZ][2]: y[-1], y[-2] at chunk start

__device__ __forceinline__ void build_frame_coeffs(const float* __restrict__ l,
                                                   float& a1, float& a2) {
  a1 = 2.0f * tanhf(l[0]) * STAB;
  float a1a = fabsf(a1);
  a2 = ((2.0f - a1a) * tanhf(l[1]) * STAB + a1a) * 0.5f;
}

// ---------------------------------------------------------------------------
// Kernel 1: interpolation of (a1,a2,b0,b1,b2) frames -> per-sample outputs via
// WMMA f32 16x16x4.  One wave computes 16 samples x 16 columns per WMMA:
//   D(16x16) = W_tile(16x4) * F(4x16),  F rows = frames fbase..fbase+3 (padded)
// grid: 8 blocks/voice * 32 voices, 256 threads (8 waves), 64 tiles/wave.
// The scatter destination per lane is loop-invariant: hoisted out of the loop.
// ---------------------------------------------------------------------------
__global__ __launch_bounds__(256) void K_interp(const float* __restrict__ logits,
                                                float* __restrict__ out) {
  __shared__ float Fr[NF * 16];            // padded frame table, cols 5..15 = 0
  const int tid = threadIdx.x;
  const int v   = blockIdx.x >> 3;

  if (tid < NF) {
    const float* l = logits + ((size_t)v * NF + tid) * 5;
    float a1, a2; build_frame_coeffs(l, a1, a2);
    float* row = Fr + tid * 16;
    row[0] = a1; row[1] = a2; row[2] = l[2]; row[3] = l[3]; row[4] = l[4];
#pragma unroll
    for (int k = 5; k < 16; ++k) row[k] = 0.0f;
  }
  __syncthreads();

  const int lane  = tid & 31;
  const int wid   = tid >> 5;
  const int n     = lane & 15;             // column (B/D) or row M (A)
  const int hi    = lane >> 4;             // lane half
  const int kb    = hi * 2;                // K pair held by this half (A & B frags)
  const int rowhi = hi * 8;                // D rows offset
  const int tileBase = (blockIdx.x & 7) * 512 + wid * 64;

  // ---- loop-invariant scatter destination per lane ----
  //   n==0 -> a_coeff3[..,1], n==1 -> a_coeff3[..,2], n==5 -> a_coeff3[..,0]=1,
  //   n in 2..4 -> b_coeff[..,n-2]; n in 6..15 inactive.
  const bool active = (n <= 5);
  const bool is_one = (n == 5);
  const size_t vbase3 = (size_t)v * NS * 3;
  float* dst;
  if      (n == 0) dst = out + OFF_A3 + vbase3 + 1;
  else if (n == 1) dst = out + OFF_A3 + vbase3 + 2;
  else if (n == 5) dst = out + OFF_A3 + vbase3 + 0;
  else             dst = out + OFF_B3 + vbase3 + (size_t)(n - 2); // n 2..4 (6..15 unused)

  for (int it = 0; it < 64; ++it) {
    const int tile = tileBase + it;
    const int t0   = tile << 4;
    const int fbase = (int)floorf((float)t0 * POS_SCALE);

    // ---- A fragment: interpolation weights W[t0+M, fbase+k], M = n ----
    const int tA   = t0 + n;
    const float pos = (float)tA * POS_SCALE;
    const float i0f = floorf(pos);
    const int   i0  = (int)i0f;
    const float fr  = pos - i0f;
    const int   i1  = min(i0 + 1, NF - 1);
    v2f a, b;
    {
      const int fA0 = fbase + kb, fA1 = fbase + kb + 1;
      a.x = ((fA0 == i0) ? (1.0f - fr) : 0.0f) + ((fA0 == i1) ? fr : 0.0f);
      a.y = ((fA1 == i0) ? (1.0f - fr) : 0.0f) + ((fA1 == i1) ? fr : 0.0f);
    }
    // ---- B fragment: frame rows fbase+kb, fbase+kb+1 (clamped), col = n ----
    {
      const int fB0 = min(fbase + kb,     NF - 1);
      const int fB1 = min(fbase + kb + 1, NF - 1);
      b.x = Fr[fB0 * 16 + n];
      b.y = Fr[fB1 * 16 + n];
    }

    v8f c = {0.f,0.f,0.f,0.f,0.f,0.f,0.f,0.f};
#if __has_builtin(__builtin_amdgcn_wmma_f32_16x16x4_f32)
    v8f d = __builtin_amdgcn_wmma_f32_16x16x4_f32(false, a, false, b,
                                                  (short)0, c, false, false);
#else
    // scalar fallback (same math, per-lane column n)
    v8f d = c;
#pragma unroll
    for (int r = 0; r < 8; ++r) {
      const int t = t0 + r + rowhi;
      const float p   = (float)t * POS_SCALE;
      const float j0f = floorf(p);
      const int   j0  = (int)j0f;
      const float f2  = p - j0f;
      const int   j1  = min(j0 + 1, NF - 1);
      d[r] = (1.0f - f2) * Fr[j0 * 16 + n] + f2 * Fr[j1 * 16 + n];
    }
#endif

    // ---- scatter: one base address per tile, constant offsets r*3 folded
    // into the store immediates; value select is a single cndmask.
    if (active) {
      float* p = dst + (size_t)(t0 + rowhi) * 3;
#pragma unroll
      for (int r = 0; r < 8; ++r) {
        p[r * 3] = is_one ? 1.0f : d[r];
      }
    }
  }
}

// ---------------------------------------------------------------------------
// Kernel 2: phase 1 — dry + envelope + chunk affine transition.
// block = voice, thread = chunk (256 samples).
// ---------------------------------------------------------------------------
__global__ __launch_bounds__(256) void K_phase1(
    const float* __restrict__ f0_hz, const float* __restrict__ dur_,
    const float* __restrict__ phase_, const float* __restrict__ logits,
    const float* __restrict__ shape_, const float* __restrict__ gain_,
    const float* __restrict__ alpha_, float* __restrict__ out,
    float* __restrict__ ws) {
  __shared__ float fc[NF * 5];
  const int v = blockIdx.x;
  const int j = threadIdx.x;
  if (j < NF) {
    const float* l = logits + ((size_t)v * NF + j) * 5;
    float a1, a2; build_frame_coeffs(l, a1, a2);
    fc[j*5+0] = a1; fc[j*5+1] = a2; fc[j*5+2] = l[2]; fc[j*5+3] = l[3]; fc[j*5+4] = l[4];
  }
  __syncthreads();

  const float f0 = f0_hz[v], ph = phase_[v], shp = shape_[v];
  const float g  = gain_[v], dur = dur_[v],  alpha = alpha_[v];
  const double uh = (double)f0 * (0.5 / 48000.0);        // half-cycle increment
  double u = (double)(j * CHUNK + 1) * uh;               // arg/(4*pi) at first sample
  const float partials = 12000.0f / (f0 * log10f(f0));
  const float k2 = HALF_PI_F * partials;
  const float s2 = 1.0f - 0.5f * shp;
  const float invdur = 1.0f / dur;

  float T00 = 1.f, T01 = 0.f, T10 = 0.f, T11 = 1.f, c0 = 0.f, c1 = 0.f;
  const size_t base = (size_t)v * NS + (size_t)j * CHUNK;

  for (int s4 = 0; s4 < CHUNK; s4 += 4) {
    float db[4], eb[4];
#pragma unroll
    for (int q = 0; q < 4; ++q) {
      const int t = j * CHUNK + s4 + q;
      // interpolated denominator coeffs
      const float pos = (float)t * POS_SCALE;
      const float i0f = floorf(pos);
      const int   i0  = (int)i0f;
      const float fr  = pos - i0f;
      const int   i1  = min(i0 + 1, NF - 1);
      const float w0  = 1.0f - fr;
      const float a1  = w0 * fc[i0*5+0] + fr * fc[i1*5+0];
      const float a2  = w0 * fc[i0*5+1] + fr * fc[i1*5+1];
      // VCO: arg = 2*pi*(t+1)*f0/SR + ph;  h = arg/2 reduced mod 2*pi
      const double uf = u - floor(u);
      const float h  = (float)uf * TWO_PI_F + 0.5f * ph;
      const float sh = __sinf(h), ch = __cosf(h);
      const float sarg = 2.0f * sh * ch;                 // sin(arg)
      const float sq   = tanhf(k2 * sarg);
      const float osc  = s2 * sq * (1.0f + shp * ch);
      // envelope
      const float tsec = (float)t * (1.0f / 48000.0f);
      float ramp = 1.0f - tsec * invdur;
      ramp = fminf(fmaxf(ramp, 0.001f), 1.0f);
      float env = exp2f(alpha * log2f(ramp));
      env = (tsec <= dur) ? env : 0.0f;
      const float x = osc * g * env;
      db[q] = x; eb[q] = env;
      // companion-matrix affine accumulation: T <- M*T, c <- M*c + (x,0)
      const float n00 = -a1 * T00 - a2 * T10;
      const float n01 = -a1 * T01 - a2 * T11;
      T10 = T00; T11 = T01; T00 = n00; T01 = n01;
      const float nc = x - a1 * c0 - a2 * c1;
      c1 = c0; c0 = nc;
      u += uh;
    }
    float4 dv; dv.x = db[0]; dv.y = db[1]; dv.z = db[2]; dv.w = db[3];
    float4 ev; ev.x = eb[0]; ev.y = eb[1]; ev.z = eb[2]; ev.w = eb[3];
    *(float4*)(out + OFF_DRY + base + s4) = dv;
    *(float4*)(out + OFF_ENV + base + s4) = ev;
  }
  float* tr = ws + WS_TRANS + ((size_t)j * BSZ + v) * 6;
  tr[0] = T00; tr[1] = T01; tr[2] = T10; tr[3] = T11; tr[4] = c0; tr[5] = c1;
}

// ---------------------------------------------------------------------------
// Kernel 3: phase 2 — sequential composition over chunks, lane = voice (1 wave)
// ---------------------------------------------------------------------------
__global__ void K_phase2(float* __restrict__ ws) {
  const int v = threadIdx.x;               // 0..31, wave32: one lane per voice
  float s0 = 0.0f, s1 = 0.0f;              // (y[-1], y[-2]) at chunk start
  for (int j = 0; j < NCHUNK; ++j) {
    float* ini = ws + WS_INIT + ((size_t)j * BSZ + v) * 2;
    ini[0] = s0; ini[1] = s1;
    const float* tr = ws + WS_TRANS + ((size_t)j * BSZ + v) * 6;
    const float n0 = tr[0] * s0 + tr[1] * s1 + tr[4];
    const float n1 = tr[2] * s0 + tr[3] * s1 + tr[5];
    s0 = n0; s1 = n1;
  }
}

// ---------------------------------------------------------------------------
// Kernel 4: phase 3 — exact IIR replay + fused FIR + tanh distortion
// ---------------------------------------------------------------------------
__global__ __launch_bounds__(256) void K_phase3(const float* __restrict__ logits,
                                                const float* __restrict__ dist_,
                                                float* __restrict__ out,
                                                const float* __restrict__ ws) {
  __shared__ float fc[NF * 5];
  const int v = blockIdx.x;
  const int j = threadIdx.x;
  if (j < NF) {
    const float* l = logits + ((size_t)v * NF + j) * 5;
    float a1, a2; build_frame_coeffs(l, a1, a2);
    fc[j*5+0] = a1; fc[j*5+1] = a2; fc[j*5+2] = l[2]; fc[j*5+3] = l[3]; fc[j*5+4] = l[4];
  }
  __syncthreads();

  const float dg = dist_[v];
  float y1 = ws[WS_INIT + ((size_t)j * BSZ + v) * 2 + 0];   // y[t-1]
  float y2 = ws[WS_INIT + ((size_t)j * BSZ + v) * 2 + 1];   // y[t-2]
  const size_t base = (size_t)v * NS + (size_t)j * CHUNK;
  const float* dryp = out + OFF_DRY + base;

  for (int s4 = 0; s4 < CHUNK; s4 += 4) {
    __builtin_prefetch(dryp + s4 + 64, 0, 0);               // global_prefetch_b8
    const float4 xq = *(const float4*)(dryp + s4);
    float xs[4]; xs[0] = xq.x; xs[1] = xq.y; xs[2] = xq.z; xs[3] = xq.w;
    float ya[4], yb[4], wv[4];
#pragma unroll
    for (int q = 0; q < 4; ++q) {
      const int t = j * CHUNK + s4 + q;
      const float pos = (float)t * POS_SCALE;
      const float i0f = floorf(pos);
      const int   i0  = (int)i0f;
      const float fr  = pos - i0f;
      const int   i1  = min(i0 + 1, NF - 1);
      const float w0  = 1.0f - fr;
      const float a1 = w0 * fc[i0*5+0] + fr * fc[i1*5+0];
      const float a2 = w0 * fc[i0*5+1] + fr * fc[i1*5+1];
      const float b0 = w0 * fc[i0*5+2] + fr * fc[i1*5+2];
      const float b1 = w0 * fc[i0*5+3] + fr * fc[i1*5+3];
      const float b2 = w0 * fc[i0*5+4] + fr * fc[i1*5+4];
      const float y   = xs[q] - a1 * y1 - a2 * y2;          // all-pole
      const float yab = b0 * y + b1 * y1 + b2 * y2;         // time-varying FIR
      ya[q] = y; yb[q] = yab; wv[q] = tanhf(yab * dg);
      y2 = y1; y1 = y;
    }
    float4 av; av.x = ya[0]; av.y = ya[1]; av.z = ya[2]; av.w = ya[3];
    float4 bv; bv.x = yb[0]; bv.y = yb[1]; bv.z = yb[2]; bv.w = yb[3];
    float4 wvv; wvv.x = wv[0]; wvv.y = wv[1]; wvv.z = wv[2]; wvv.w = wv[3];
    *(float4*)(out + OFF_YA  + base + s4) = av;
    *(float4*)(out + OFF_YAB + base + s4) = bv;
    *(float4*)(out + OFF_WET + base + s4) = wvv;
  }
}

// ---------------------------------------------------------------------------
extern "C" void kernel_launch(void* const* d_in, const int* in_sizes, int n_in,
                              void* d_out, int out_size, void* d_ws, size_t ws_size,
                              hipStream_t stream) {
  (void)in_sizes; (void)n_in; (void)out_size; (void)ws_size;
  const float* f0    = (const float*)d_in[0];
  const float* dur   = (const float*)d_in[1];
  const float* ph    = (const float*)d_in[2];
  const float* lg    = (const float*)d_in[3];
  const float* shp   = (const float*)d_in[4];
  const float* gain  = (const float*)d_in[5];
  const float* dist  = (const float*)d_in[6];
  const float* alpha = (const float*)d_in[7];
  float* out = (float*)d_out;
  float* ws  = (float*)d_ws;

  K_interp<<<dim3(BSZ * 8), dim3(256), 0, stream>>>(lg, out);
  K_phase1<<<dim3(BSZ), dim3(256), 0, stream>>>(f0, dur, ph, lg, shp, gain, alpha, out, ws);
  K_phase2<<<dim3(1), dim3(32), 0, stream>>>(ws);
  K_phase3<<<dim3(BSZ), dim3(256), 0, stream>>>(lg, dist, out, ws);
}